// TensorProductUniform3x1d_69295002354281
// MI455X (gfx1250) — compile-verified
//
#include <hip/hip_runtime.h>
#include <hip/hip_bf16.h>
#include <stdint.h>

// Problem constants (from the reference descriptor)
#define S0U        512     // S0*U floats per x0 row
#define S1U        512     // S1*U floats per x1 row
#define S2U        1024    // S2*U floats per out row
#define NSEG_OUT   8
#define NPATH      16
#define U_CHUNKS   32      // 128 floats / 4 per lane = 32 lanes
#define WAVES_PER_BLOCK 8
#define BLOCK_THREADS   (WAVES_PER_BLOCK * 32)

// Native clang vector type: used ONLY at the final store site so
// __builtin_nontemporal_store emits b128 th:NT. The compute loop stays on
// HIP float4 (struct) so SROA keeps the accumulators as scalar floats and
// the uniform-branch accumulate doesn't grow vector phis (round-3 blowup).
typedef float v4f __attribute__((ext_vector_type(4)));

// ---- CDNA5 async global->LDS copy (ASYNCcnt-tracked), per ISA 15.18.3 ----
// Non-temporal: inputs are streamed exactly once (820MB/pass vs 192MB L2).
__device__ __forceinline__ void async_g2l_b128_nt(unsigned lds_byte_addr, const void* gaddr) {
    // vdst = VGPR holding LDS byte address, vaddr = 64-bit global address, saddr = off
    asm volatile("global_load_async_to_lds_b128 %0, %1, off th:TH_LOAD_NT"
                 :: "v"(lds_byte_addr), "v"(gaddr)
                 : "memory");
}

__device__ __forceinline__ void wait_asynccnt0() {
#if __has_builtin(__builtin_amdgcn_s_wait_asynccnt)
    __builtin_amdgcn_s_wait_asynccnt(0);
#else
    asm volatile("s_wait_asynccnt 0" ::: "memory");
#endif
}

__global__ __launch_bounds__(BLOCK_THREADS)
void TensorProductUniform3x1d_kernel(const float* __restrict__ x0,
                                     const float* __restrict__ x1,
                                     const int*   __restrict__ pidx,   // [P,3]
                                     const float* __restrict__ pcoef,  // [P]
                                     float* __restrict__ out,
                                     int Z)
{
    // Per-wave 4KB slab: [0..511] = x0 row, [512..1023] = x1 row
    __shared__ float smem[WAVES_PER_BLOCK * (S0U + S1U)];

    const int lane = threadIdx.x & 31;
    const int wave = threadIdx.x >> 5;
    // z is wave-uniform; readfirstlane lets the guard compile to s_cmp/s_cbranch
    // (no v_cmp + exec-mask save).
    const int z = __builtin_amdgcn_readfirstlane(blockIdx.x * WAVES_PER_BLOCK + wave);
    if (z >= Z) return;                       // scalar, wave-uniform exit

    float* wbase = &smem[wave * (S0U + S1U)];
    const float* a_g = x0 + (size_t)z * S0U;
    const float* b_g = x1 + (size_t)z * S1U;

    // ---- Stage both input rows into LDS asynchronously (8 x b128 per lane) ----
#pragma unroll
    for (int q = 0; q < 4; ++q) {
        const int j = q * 32 + lane;          // float4 chunk 0..127 within the row
        const unsigned lA = (unsigned)(uintptr_t)(wbase + 4 * j);
        const unsigned lB = (unsigned)(uintptr_t)(wbase + S0U + 4 * j);
        async_g2l_b128_nt(lA, a_g + 4 * j);
        async_g2l_b128_nt(lB, b_g + 4 * j);
    }

    // ---- Path table: wave-uniform -> SGPRs (scalar compares/branches below) ----
    int   pi0[NPATH], pi1[NPATH], pi2[NPATH];
    float pcf[NPATH];
#pragma unroll
    for (int p = 0; p < NPATH; ++p) {
        pi0[p] = __builtin_amdgcn_readfirstlane(pidx[3 * p + 0]);
        pi1[p] = __builtin_amdgcn_readfirstlane(pidx[3 * p + 1]);
        pi2[p] = __builtin_amdgcn_readfirstlane(pidx[3 * p + 2]);
        pcf[p] = pcoef[p];
    }

    wait_asynccnt0();                          // LDS slab is now valid for this wave

    const float4* aS = (const float4*)wbase;            // 4 segs x 32 chunks
    const float4* bS = (const float4*)(wbase + S0U);    // 4 segs x 32 chunks

    float4 acc[NSEG_OUT];
#pragma unroll
    for (int s = 0; s < NSEG_OUT; ++s) acc[s] = make_float4(0.f, 0.f, 0.f, 0.f);

#pragma unroll
    for (int p = 0; p < NPATH; ++p) {
        const float4 a = aS[pi0[p] * U_CHUNKS + lane];  // ds_load_b128 gather
        const float4 b = bS[pi1[p] * U_CHUNKS + lane];  // ds_load_b128 gather
        const float  c = pcf[p];
        float4 ca;                                      // fold coeff into a
        ca.x = c * a.x;  ca.y = c * a.y;  ca.z = c * a.z;  ca.w = c * a.w;
        const int oseg = pi2[p];                        // SGPR -> uniform branch
#pragma unroll
        for (int s = 0; s < NSEG_OUT; ++s) {
            if (oseg == s) {
                acc[s].x = fmaf(ca.x, b.x, acc[s].x);
                acc[s].y = fmaf(ca.y, b.y, acc[s].y);
                acc[s].z = fmaf(ca.z, b.z, acc[s].z);
                acc[s].w = fmaf(ca.w, b.w, acc[s].w);
            }
        }
    }

    // ---- Coalesced NT b128 stores; untouched segments correctly write zeros ----
    // Convert float4 -> native vector only here, outside the branchy region.
    v4f* og = (v4f*)(out + (size_t)z * S2U);
#pragma unroll
    for (int s = 0; s < NSEG_OUT; ++s) {
        v4f vs;
        vs.x = acc[s].x;  vs.y = acc[s].y;  vs.z = acc[s].z;  vs.w = acc[s].w;
        __builtin_nontemporal_store(vs, &og[s * U_CHUNKS + lane]);
    }
}

extern "C" void kernel_launch(void* const* d_in, const int* in_sizes, int n_in,
                              void* d_out, int out_size, void* d_ws, size_t ws_size,
                              hipStream_t stream) {
    const float* x0    = (const float*)d_in[0];
    const float* x1    = (const float*)d_in[1];
    const int*   pidx  = (const int*)  d_in[2];
    const float* pcoef = (const float*)d_in[3];
    float* out = (float*)d_out;

    const int Z = in_sizes[0] / S0U;
    const int blocks = (Z + WAVES_PER_BLOCK - 1) / WAVES_PER_BLOCK;

    TensorProductUniform3x1d_kernel<<<blocks, BLOCK_THREADS, 0, stream>>>(
        x0, x1, pidx, pcoef, out, Z);
}